// TriangleCollisionLoss_20847771254922
// MI455X (gfx1250) — compile-verified
//
#include <hip/hip_runtime.h>
#include <math.h>

// ---------------------------------------------------------------------------
// TriangleCollisionLoss for MI455X (gfx1250, wave32).
//  Phase 2 (dominant) uses V_WMMA_F32_16X16X4_F32: score(i,j) = sq_j - 2*ci.cj
//  (per-row monotone with squared distance), A = [cx,cy,cz,1], B = [-2c, sq].
// ---------------------------------------------------------------------------

typedef __attribute__((ext_vector_type(2))) float v2f;
typedef __attribute__((ext_vector_type(8))) float v8f;

#define KNN_K     50          // min(K, F-1) for this problem
#define NBR_STR   64          // padded neighbor-list stride
#define TILE_ROWS 16
#define TILE_COLS 128         // 8 waves * 16 columns per block iteration
#define STAGE_LD  132         // padded leading dim (bank-conflict dodge)
#define POOL_CAP  384
#define POOL_STR  400         // padded (400 % 64 == 16 -> half-waves disjoint banks)

#define EPS_F     1e-8f
#define COLL_THR  1e-10f
#define NORM_THR  0.99f

// ---------------------------------------------------------------------------
// Kernel 1: per-face geometry precompute
// ---------------------------------------------------------------------------
__global__ __launch_bounds__(256)
void precompute_kernel(const float* __restrict__ verts,
                       const int*   __restrict__ faces,
                       float4* __restrict__ cB,   float4* __restrict__ cent4,
                       float4* __restrict__ nun4, float4* __restrict__ nrm4,
                       float4* __restrict__ v0a,  float4* __restrict__ v1a,
                       float4* __restrict__ v2a,  int F) {
  const int i = blockIdx.x * blockDim.x + threadIdx.x;
  if (i >= F) return;
  const int i0 = faces[3 * i + 0], i1 = faces[3 * i + 1], i2 = faces[3 * i + 2];
  const float x0 = verts[3 * i0], y0 = verts[3 * i0 + 1], z0 = verts[3 * i0 + 2];
  const float x1 = verts[3 * i1], y1 = verts[3 * i1 + 1], z1 = verts[3 * i1 + 2];
  const float x2 = verts[3 * i2], y2 = verts[3 * i2 + 1], z2 = verts[3 * i2 + 2];

  const float e1x = x1 - x0, e1y = y1 - y0, e1z = z1 - z0;
  const float e2x = x2 - x0, e2y = y2 - y0, e2z = z2 - z0;
  const float nx = e1y * e2z - e1z * e2y;
  const float ny = e1z * e2x - e1x * e2z;
  const float nz = e1x * e2y - e1y * e2x;
  const float nl = sqrtf(nx * nx + ny * ny + nz * nz) + EPS_F;

  const float cx = (x0 + x1 + x2) * (1.0f / 3.0f);
  const float cy = (y0 + y1 + y2) * (1.0f / 3.0f);
  const float cz = (z0 + z1 + z2) * (1.0f / 3.0f);
  const float sq = cx * cx + cy * cy + cz * cz;

  cB[i]    = make_float4(-2.0f * cx, -2.0f * cy, -2.0f * cz, sq);
  cent4[i] = make_float4(cx, cy, cz, 0.0f);
  nun4[i]  = make_float4(nx, ny, nz, 0.0f);
  nrm4[i]  = make_float4(nx / nl, ny / nl, nz / nl, 0.0f);
  v0a[i]   = make_float4(x0, y0, z0, 0.0f);
  v1a[i]   = make_float4(x1, y1, z1, 0.0f);
  v2a[i]   = make_float4(x2, y2, z2, 0.0f);
}

// ---------------------------------------------------------------------------
// Half-wave (16-lane team) selection: find the KNN_K smallest of pool[0..n)
// by repeated lexicographic-min extraction on the key (value, pool_index).
// Extraction is read-only on the pool (each round only considers keys
// strictly greater than the last extracted key), so no volatile / no LDS
// write hazards: everything lowers to plain ds_load + lane shuffles.
// Afterwards the kept elements are compacted to the pool front (ascending).
// xor-shuffle masks <= 8 stay inside a 16-lane half, so the two teams that
// share one wave never interfere, even under divergence.
// Returns the max (i.e. last extracted) kept value.
// ---------------------------------------------------------------------------
#define POOL_J_NONE 0x7fffffff

__device__ __forceinline__
float team_select(float* pd, int* pi, int n, int slot,
                  float kd[4], int ki[4]) {
  float lastv = -INFINITY;
  int   lastj = -1;
  for (int e = 0; e < KNN_K; ++e) {
    float bv = INFINITY;
    int   bj = POOL_J_NONE;
    for (int j = slot; j < n; j += 16) {
      const float v    = pd[j];
      const bool  elig = (v > lastv) || (v == lastv && j > lastj);
      if (elig && (v < bv || (v == bv && j < bj))) { bv = v; bj = j; }
    }
#pragma unroll
    for (int off = 8; off >= 1; off >>= 1) {
      const float ov = __shfl_xor(bv, off, 32);
      const int   oj = __shfl_xor(bj, off, 32);
      if (ov < bv || (ov == bv && oj < bj)) { bv = ov; bj = oj; }
    }
    if ((e & 15) == slot) {
      kd[e >> 4] = bv;
      ki[e >> 4] = (bj != POOL_J_NONE) ? pi[bj] : -1;
    }
    lastv = bv;
    lastj = bj;
  }
  // compact kept elements to the pool front (reads all done above)
#pragma unroll
  for (int r = 0; r < 4; ++r) {
    const int e = slot + r * 16;
    if (e < KNN_K) { pd[e] = kd[r]; pi[e] = ki[r]; }
  }
  return lastv;
}

// ---------------------------------------------------------------------------
// Kernel 2: fused WMMA score tiles + threshold-filtered 50-NN per face.
// Block = 256 threads (8 waves) owns 16 rows; 157 iterations of 128 columns.
// ---------------------------------------------------------------------------
__global__ __launch_bounds__(256)
void knn_kernel(const float4* __restrict__ cent4,
                const float4* __restrict__ cB,
                int* __restrict__ nbr, int F) {
  __shared__ float stage[TILE_ROWS * STAGE_LD];
  __shared__ float pool_d[TILE_ROWS * POOL_STR];
  __shared__ int   pool_i[TILE_ROWS * POOL_STR];
  __shared__ int   cnt[TILE_ROWS];
  __shared__ float thr[TILE_ROWS];

  const int tid  = threadIdx.x;
  const int lane = tid & 31;
  const int wave = tid >> 5;
  const int row  = tid >> 4;      // team id == local row 0..15
  const int slot = tid & 15;
  const int r0   = blockIdx.x * TILE_ROWS;

  if (tid < TILE_ROWS) { cnt[tid] = 0; thr[tid] = INFINITY; }

  // A fragment (identical in every wave): A[m][:] = [cx, cy, cz, 1]
  // f32 16x4 layout: lanes 0-15 hold (K0,K1), lanes 16-31 hold (K2,K3).
  const int    am   = lane & 15;
  const int    arow = min(r0 + am, F - 1);
  const float4 ac   = cent4[arow];
  v2f afrag;
  afrag.x = (lane < 16) ? ac.x : ac.z;
  afrag.y = (lane < 16) ? ac.y : 1.0f;

  __syncthreads();

  const int rowface = r0 + row;
  float*    pd = &pool_d[row * POOL_STR];
  int*      pi = &pool_i[row * POOL_STR];
  float kd[4]; int ki[4];

  const int nIter = (F + TILE_COLS - 1) / TILE_COLS;
  for (int it = 0; it < nIter; ++it) {
    const int base = it * TILE_COLS;

    // Compact before the pool can overflow this iteration (<=128 new appends).
    if (rowface < F && cnt[row] >= POOL_CAP - TILE_COLS) {
      const int   n = min(cnt[row], POOL_CAP);
      const float t = team_select(pd, pi, n, slot, kd, ki);
      if (slot == 0) { cnt[row] = KNN_K; thr[row] = t; }
    }
    __syncthreads();

    // WMMA score tile: wave w covers columns [base + 16w, base + 16w + 16).
    {
      const int    n    = lane & 15;
      const int    col  = base + wave * 16 + n;
      const float4 bc   = cB[min(col, F - 1)];   // out-of-range filtered later
      v2f bfrag;
      bfrag.x = (lane < 16) ? bc.x : bc.z;       // rows K0/K1 vs K2/K3 of B
      bfrag.y = (lane < 16) ? bc.y : bc.w;
      v8f c = {};
      c = __builtin_amdgcn_wmma_f32_16x16x4_f32(
          /*neg_a=*/false, afrag, /*neg_b=*/false, bfrag,
          /*c_mod=*/(short)0, c, /*reuse_a=*/false, /*reuse_b=*/false);
      const int m0 = (lane < 16) ? 0 : 8;        // C layout: vgpr r -> row r / r+8
      const int sc = wave * 16 + n;
#pragma unroll
      for (int r = 0; r < 8; ++r)
        stage[(m0 + r) * STAGE_LD + sc] = c[r];
    }
    __syncthreads();

    // Row teams filter their 128 staged candidates against the threshold.
    if (rowface < F) {
      const float t = thr[row];
#pragma unroll
      for (int q = 0; q < TILE_COLS / 16; ++q) {
        const int   sc  = slot + q * 16;
        const int   col = base + sc;
        const float s   = stage[row * STAGE_LD + sc];
        if (col < F && col != rowface && s < t) {
          const int pos = atomicAdd(&cnt[row], 1);
          if (pos < POOL_CAP) { pd[pos] = s; pi[pos] = col; }
        }
      }
    }
    __syncthreads();
  }

  // Final: extract the 50 nearest, emit neighbor indices (order irrelevant
  // downstream; count is a sum over neighbors).
  if (rowface < F) {
    const int n = min(cnt[row], POOL_CAP);
    (void)team_select(pd, pi, n, slot, kd, ki);
#pragma unroll
    for (int r = 0; r < 4; ++r) {
      const int e = slot + r * 16;
      if (e < KNN_K) nbr[rowface * NBR_STR + e] = ki[r];
    }
  }
}

// ---------------------------------------------------------------------------
// Kernel 3: collision test per (face, neighbor); per-face weighted count.
// ---------------------------------------------------------------------------
__global__ __launch_bounds__(64)
void collide_kernel(const float4* __restrict__ cent4, const float4* __restrict__ nun4,
                    const float4* __restrict__ nrm4,  const float4* __restrict__ v0a,
                    const float4* __restrict__ v1a,   const float4* __restrict__ v2a,
                    const int* __restrict__ faces,    const float* __restrict__ prob,
                    const int* __restrict__ nbr,      float* __restrict__ part, int F) {
  const int i = blockIdx.x;
  __shared__ int scnt;
  if (threadIdx.x == 0) scnt = 0;
  __syncthreads();

  const int t = threadIdx.x;
  if (t < KNN_K) {
    const int j = nbr[i * NBR_STR + t];
    bool coll = false;
    if (j >= 0 && j < F) {
      const float4 ni = nrm4[i], nj = nrm4[j];
      const float  ndot = fabsf(ni.x * nj.x + ni.y * nj.y + ni.z * nj.z);
      bool inter;
      if (ndot > NORM_THR) {
        const float4 ci = cent4[i], cj = cent4[j];
        const float dx = ci.x - cj.x, dy = ci.y - cj.y, dz = ci.z - cj.z;
        inter = sqrtf(dx * dx + dy * dy + dz * dz) < COLL_THR;
      } else {
        const float4 nui = nun4[i], nuj = nun4[j];
        const float4 a0 = v0a[i], a1 = v1a[i], a2 = v2a[i];
        const float4 b0 = v0a[j], b1 = v1a[j], b2 = v2a[j];
        const float dA0 = (b0.x - a0.x) * nui.x + (b0.y - a0.y) * nui.y + (b0.z - a0.z) * nui.z;
        const float dA1 = (b1.x - a0.x) * nui.x + (b1.y - a0.y) * nui.y + (b1.z - a0.z) * nui.z;
        const float dA2 = (b2.x - a0.x) * nui.x + (b2.y - a0.y) * nui.y + (b2.z - a0.z) * nui.z;
        const bool condA = (dA0 * dA1 <= 0.0f) || (dA0 * dA2 <= 0.0f) || (dA1 * dA2 <= 0.0f);
        const float dB0 = (a0.x - b0.x) * nuj.x + (a0.y - b0.y) * nuj.y + (a0.z - b0.z) * nuj.z;
        const float dB1 = (a1.x - b0.x) * nuj.x + (a1.y - b0.y) * nuj.y + (a1.z - b0.z) * nuj.z;
        const float dB2 = (a2.x - b0.x) * nuj.x + (a2.y - b0.y) * nuj.y + (a2.z - b0.z) * nuj.z;
        const bool condB = (dB0 * dB1 <= 0.0f) || (dB0 * dB2 <= 0.0f) || (dB1 * dB2 <= 0.0f);
        inter = condA && condB;
      }
      // Shared-vertex adjacency with the reference's "first occurrence" dedup.
      const int f0 = faces[3 * i], f1 = faces[3 * i + 1], f2 = faces[3 * i + 2];
      const int g0 = faces[3 * j], g1 = faces[3 * j + 1], g2 = faces[3 * j + 2];
      int sh = 0;
      sh += ((f0 == g0) || (f0 == g1) || (f0 == g2)) ? 1 : 0;
      sh += ((f1 != f0) && ((f1 == g0) || (f1 == g1) || (f1 == g2))) ? 1 : 0;
      sh += ((f2 != f0) && (f2 != f1) && ((f2 == g0) || (f2 == g1) || (f2 == g2))) ? 1 : 0;
      coll = inter && (sh < 2);
    }
    if (coll) atomicAdd(&scnt, 1);   // integer LDS atomic: deterministic
  }
  __syncthreads();
  if (threadIdx.x == 0) part[i] = prob[i] * (float)scnt;
}

// ---------------------------------------------------------------------------
// Kernel 4: fixed-order deterministic reduction of per-face partials.
// ---------------------------------------------------------------------------
__global__ __launch_bounds__(256)
void reduce_kernel(const float* __restrict__ part, float* __restrict__ out, int F) {
  __shared__ float s[256];
  float acc = 0.0f;
  for (int i = threadIdx.x; i < F; i += 256) acc += part[i];
  s[threadIdx.x] = acc;
  __syncthreads();
  for (int off = 128; off > 0; off >>= 1) {
    if (threadIdx.x < off) s[threadIdx.x] += s[threadIdx.x + off];
    __syncthreads();
  }
  if (threadIdx.x == 0) out[0] = s[0];
}

// ---------------------------------------------------------------------------
extern "C" void kernel_launch(void* const* d_in, const int* in_sizes, int n_in,
                              void* d_out, int out_size, void* d_ws, size_t ws_size,
                              hipStream_t stream) {
  const float* verts = (const float*)d_in[0];
  const int*   faces = (const int*)d_in[1];
  const float* prob  = (const float*)d_in[2];
  const int F = in_sizes[1] / 3;
  (void)n_in; (void)out_size; (void)ws_size;

  char*  ws  = (char*)d_ws;
  size_t off = 0;
  auto carve = [&](size_t bytes) -> void* {
    void* p = (void*)(ws + off);
    off += (bytes + 255) & ~(size_t)255;
    return p;
  };
  float4* cB    = (float4*)carve((size_t)F * sizeof(float4));
  float4* cent4 = (float4*)carve((size_t)F * sizeof(float4));
  float4* nun4  = (float4*)carve((size_t)F * sizeof(float4));
  float4* nrm4  = (float4*)carve((size_t)F * sizeof(float4));
  float4* v0a   = (float4*)carve((size_t)F * sizeof(float4));
  float4* v1a   = (float4*)carve((size_t)F * sizeof(float4));
  float4* v2a   = (float4*)carve((size_t)F * sizeof(float4));
  int*    nbr   = (int*)  carve((size_t)F * NBR_STR * sizeof(int));
  float*  part  = (float*)carve((size_t)F * sizeof(float));

  precompute_kernel<<<(F + 255) / 256, 256, 0, stream>>>(
      verts, faces, cB, cent4, nun4, nrm4, v0a, v1a, v2a, F);
  knn_kernel<<<(F + TILE_ROWS - 1) / TILE_ROWS, 256, 0, stream>>>(
      cent4, cB, nbr, F);
  collide_kernel<<<F, 64, 0, stream>>>(
      cent4, nun4, nrm4, v0a, v1a, v2a, faces, prob, nbr, part, F);
  reduce_kernel<<<1, 256, 0, stream>>>(part, (float*)d_out, F);
}